// LoTDEncoding_13537736917451
// MI455X (gfx1250) — compile-verified
//
#include <hip/hip_runtime.h>
#include <cstdint>
#include <cstddef>

// ---------------------------------------------------------------------------
// LoTD multi-level dense-grid trilinear interpolation for MI455X (gfx1250).
// Levels: res {16,32,64,128,256}, feats {4,4,4,2,2}  -> 16 floats / point.
// Param table = 148.6 MB -> fully L2 (192MB) resident: gathers are L2 hits.
// HBM floor: ~12MB in + 64MB out + 149MB params ~= 10us @ 23.3 TB/s.
// Level 0 (64 KB) is staged into LDS via async global->LDS loads (ASYNCcnt),
// big levels (128^3, 256^3) are prefetched (global_prefetch_b8), and the
// 64MB output is written with non-temporal 128-bit stores so it does not
// evict the L2-resident parameter table.
// ---------------------------------------------------------------------------

typedef float v4f __attribute__((ext_vector_type(4)));
typedef float v2f __attribute__((ext_vector_type(2)));
typedef int   v4i __attribute__((ext_vector_type(4)));

// Level offsets in floats (cumulative sizes of R^3 * F)
#define OFF_L1 16384
#define OFF_L2 147456
#define OFF_L3 1196032
#define OFF_L4 5390336

#define L0_FLOATS 16384   // 16^3 * 4 floats = 64 KB
#define BLOCK 256
#define PPT 4             // points per thread (amortizes LDS staging 4x)

static __device__ __forceinline__ int clampi(int v, int hi) {
    v = v < 0 ? 0 : v;
    return v > hi ? hi : v;
}

// Trilinear interpolation, F = 4 features (float4 gathers).
template <int R>
static __device__ __forceinline__ v4f interp_f4(const float* __restrict__ base,
                                                float x, float y, float z) {
    const float fx = x * (float)(R - 1);
    const float fy = y * (float)(R - 1);
    const float fz = z * (float)(R - 1);
    const int ix = clampi((int)floorf(fx), R - 2);
    const int iy = clampi((int)floorf(fy), R - 2);
    const int iz = clampi((int)floorf(fz), R - 2);
    const float wx = fx - (float)ix;
    const float wy = fy - (float)iy;
    const float wz = fz - (float)iz;
    const v4f* __restrict__ g = (const v4f*)base;
    const int b00 = (ix * R + iy) * R + iz;   // (x0,y0,z0)
    const int b01 = b00 + R;                  // (x0,y1,z0)
    const int b10 = b00 + R * R;              // (x1,y0,z0)
    const int b11 = b10 + R;                  // (x1,y1,z0)
    // Issue all 8 gathers back-to-back so they batch under one loadcnt wait.
    const v4f c000 = g[b00], c001 = g[b00 + 1];
    const v4f c010 = g[b01], c011 = g[b01 + 1];
    const v4f c100 = g[b10], c101 = g[b10 + 1];
    const v4f c110 = g[b11], c111 = g[b11 + 1];
    const v4f c00 = c000 * (1.f - wz) + c001 * wz;
    const v4f c01 = c010 * (1.f - wz) + c011 * wz;
    const v4f c10 = c100 * (1.f - wz) + c101 * wz;
    const v4f c11 = c110 * (1.f - wz) + c111 * wz;
    const v4f c0 = c00 * (1.f - wy) + c01 * wy;
    const v4f c1 = c10 * (1.f - wy) + c11 * wy;
    return c0 * (1.f - wx) + c1 * wx;
}

// Precomputed corner pointers + weights for an F=2 level, shared between the
// prefetch pass and the gather pass (avoids duplicated address math).
struct C2 {
    const v2f* p00;   // (x0,y0,z0)
    const v2f* p01;   // (x0,y1,z0)
    const v2f* p10;   // (x1,y0,z0)
    const v2f* p11;   // (x1,y1,z0)
    float wx, wy, wz;
};

template <int R>
static __device__ __forceinline__ C2 corners_f2(const float* __restrict__ base,
                                                float x, float y, float z) {
    const float fx = x * (float)(R - 1);
    const float fy = y * (float)(R - 1);
    const float fz = z * (float)(R - 1);
    const int ix = clampi((int)floorf(fx), R - 2);
    const int iy = clampi((int)floorf(fy), R - 2);
    const int iz = clampi((int)floorf(fz), R - 2);
    const v2f* g = (const v2f*)base;
    const int b00 = (ix * R + iy) * R + iz;
    C2 c;
    c.p00 = g + b00;
    c.p01 = g + b00 + R;
    c.p10 = g + b00 + R * R;
    c.p11 = g + b00 + R * R + R;
    c.wx = fx - (float)ix;
    c.wy = fy - (float)iy;
    c.wz = fz - (float)iz;
    return c;
}

static __device__ __forceinline__ void prefetch_f2(const C2& c) {
    __builtin_prefetch((const void*)c.p00, 0, 3);
    __builtin_prefetch((const void*)c.p01, 0, 3);
    __builtin_prefetch((const void*)c.p10, 0, 3);
    __builtin_prefetch((const void*)c.p11, 0, 3);
}

static __device__ __forceinline__ v2f interp_f2(const C2& c) {
    const v2f c000 = c.p00[0], c001 = c.p00[1];
    const v2f c010 = c.p01[0], c011 = c.p01[1];
    const v2f c100 = c.p10[0], c101 = c.p10[1];
    const v2f c110 = c.p11[0], c111 = c.p11[1];
    const v2f c00 = c000 * (1.f - c.wz) + c001 * c.wz;
    const v2f c01 = c010 * (1.f - c.wz) + c011 * c.wz;
    const v2f c10 = c100 * (1.f - c.wz) + c101 * c.wz;
    const v2f c11 = c110 * (1.f - c.wz) + c111 * c.wz;
    const v2f c0 = c00 * (1.f - c.wy) + c01 * c.wy;
    const v2f c1 = c10 * (1.f - c.wy) + c11 * c.wy;
    return c0 * (1.f - c.wx) + c1 * c.wx;
}

#if __has_builtin(__builtin_amdgcn_global_load_async_to_lds_b128)
#define LOTD_ASYNC_LDS 1
#else
#define LOTD_ASYNC_LDS 0
#endif

__global__ __launch_bounds__(BLOCK) void lotd_encoding_kernel(
    const float* __restrict__ in,      // [N,3] in [-1,1]
    const float* __restrict__ params,  // [38944768]
    float* __restrict__ out,           // [N,16]
    int n) {
    // ---- Stage level-0 grid (64 KB) into LDS --------------------------------
    // Fully unrolled constant trip count: 16 straight async issues per lane,
    // no divergence-masked loop around the builtin.
    __shared__ __align__(16) float lds0[L0_FLOATS];
#if LOTD_ASYNC_LDS
    {
        typedef __attribute__((address_space(3))) v4i* l4i_t;
        v4i* s = (v4i*)params + threadIdx.x;
        l4i_t d = (l4i_t)(v4f*)lds0 + threadIdx.x;
#pragma unroll
        for (int k = 0; k < L0_FLOATS / 4 / BLOCK; ++k) {  // 16 iterations
            __builtin_amdgcn_global_load_async_to_lds_b128(
                s + k * BLOCK, d + k * BLOCK, 0, 0);
        }
        asm volatile("s_wait_asynccnt 0" ::: "memory");
    }
#else
    {
        v4f* __restrict__ d = (v4f*)lds0 + threadIdx.x;
        const v4f* __restrict__ s = (const v4f*)params + threadIdx.x;
#pragma unroll
        for (int k = 0; k < L0_FLOATS / 4 / BLOCK; ++k)
            d[k * BLOCK] = s[k * BLOCK];
    }
#endif
    __syncthreads();

    const int pbase = blockIdx.x * (BLOCK * PPT) + threadIdx.x;

#pragma unroll
    for (int k = 0; k < PPT; ++k) {
        const int p = pbase + k * BLOCK;
        if (p >= n) continue;

        // ---- Load point, map [-1,1] -> [0,1] --------------------------------
        const float* ip = in + 3 * (size_t)p;
        const float x = fminf(fmaxf(ip[0] * 0.5f + 0.5f, 0.f), 1.f);
        const float y = fminf(fmaxf(ip[1] * 0.5f + 0.5f, 0.f), 1.f);
        const float z = fminf(fmaxf(ip[2] * 0.5f + 0.5f, 0.f), 1.f);

        // Corner addresses for the two largest (highest-latency) levels,
        // computed once and shared by prefetch + gather.
        const C2 c4 = corners_f2<256>(params + OFF_L4, x, y, z);
        const C2 c3 = corners_f2<128>(params + OFF_L3, x, y, z);
        prefetch_f2(c4);
        prefetch_f2(c3);

        // ---- Interpolate all 5 levels ---------------------------------------
        const v4f f1 = interp_f4<32>(params + OFF_L1, x, y, z);  // L2 gather
        const v4f f2 = interp_f4<64>(params + OFF_L2, x, y, z);  // L2 gather
        const v4f f0 = interp_f4<16>(lds0, x, y, z);             // LDS gather
        const v2f f3 = interp_f2(c3);                            // prefetched
        const v2f f4 = interp_f2(c4);                            // prefetched

        // ---- Non-temporal 128-bit stores: 64 contiguous bytes per lane ------
        v4f* op = (v4f*)(out + (size_t)p * 16);
        const v4f o3 = {f3.x, f3.y, f4.x, f4.y};
        __builtin_nontemporal_store(f0, op + 0);
        __builtin_nontemporal_store(f1, op + 1);
        __builtin_nontemporal_store(f2, op + 2);
        __builtin_nontemporal_store(o3, op + 3);
    }
}

extern "C" void kernel_launch(void* const* d_in, const int* in_sizes, int n_in,
                              void* d_out, int out_size, void* d_ws, size_t ws_size,
                              hipStream_t stream) {
    (void)n_in; (void)out_size; (void)d_ws; (void)ws_size;
    const float* in = (const float*)d_in[0];
    const float* params = (const float*)d_in[1];
    float* out = (float*)d_out;
    const int n = in_sizes[0] / 3;
    const int grid = (n + BLOCK * PPT - 1) / (BLOCK * PPT);
    lotd_encoding_kernel<<<grid, BLOCK, 0, stream>>>(in, params, out, n);
}